// DeConvAfterDownSampling_83193516523865
// MI455X (gfx1250) — compile-verified
//
#include <hip/hip_runtime.h>
#include <hip/hip_bf16.h>

typedef __attribute__((ext_vector_type(2))) float v2f;
typedef __attribute__((ext_vector_type(8))) float v8f;

// Problem constants (fixed by the reference setup)
#define NROWS 32768   // B*C*H = 8*64*64
#define WSUM  64      // width summed per row
#define FNUM  64      // filters
#define PNUM  25      // R*C = 5*5

// ---------------------------------------------------------------------------
// Kernel 1: s[n] = sum_w x[n*64 + w]   (one wave32 per row, coalesced)
// ---------------------------------------------------------------------------
__global__ void row_sum_kernel(const float* __restrict__ x,
                               float* __restrict__ s) {
    const int wave = (blockIdx.x * blockDim.x + threadIdx.x) >> 5;
    const int lane = threadIdx.x & 31;
    const float* row = x + (size_t)wave * WSUM;
    float v = row[lane] + row[lane + 32];
    // full wave32 butterfly reduction
    #pragma unroll
    for (int off = 16; off > 0; off >>= 1)
        v += __shfl_xor(v, off, 32);
    if (lane == 0)
        s[wave] = v;
}

// ---------------------------------------------------------------------------
// Kernel 2: out[f,n,p] = relu(s[n]*Wf[f,p] + b[f]) via V_WMMA_F32_16X16X4_F32
//   A (16x4):  A[m,0] = s[nb+m], K=1..3 zero
//   B (4x16):  B[0,j] = Wf[f, pb+j], K=1..3 zero
//   C (16x16): b[f] broadcast  ->  D = s*W + b in one WMMA, bias fused.
// One wave per 16(n) x 16(p) tile; p-tile 1 has only 9 valid columns (P=25).
// ---------------------------------------------------------------------------
__global__ void outer_wmma_kernel(const float* __restrict__ s,
                                  const float* __restrict__ Wf,
                                  const float* __restrict__ bias,
                                  float* __restrict__ out) {
    const int wave = blockIdx.x * 8 + (threadIdx.x >> 5);
    const int lane = threadIdx.x & 31;

    // tile id: f = wave>>12, ntile = (wave>>1)&2047, ptile = wave&1
    const int f  = wave >> 12;
    const int nb = ((wave >> 1) & 2047) << 4;   // n-tile base
    const int pb = (wave & 1) << 4;             // p-tile base (0 or 16)

    const int  l16 = lane & 15;
    const bool lo  = lane < 16;

    // A: lanes 0-15 carry {s[nb+m], 0} (K=0,1); lanes 16-31 carry {0,0} (K=2,3)
    const float sval = s[nb + l16];
    v2f a;
    a.x = lo ? sval : 0.0f;
    a.y = 0.0f;

    // B: lanes 0-15 VGPR0 = W[f, pb+j] (K=0,N=j); all other slots zero
    const int   pcol  = pb + l16;
    const bool  valid = (pcol < PNUM);
    const float wval  = valid ? Wf[f * PNUM + pcol] : 0.0f;
    v2f bmat;
    bmat.x = lo ? wval : 0.0f;
    bmat.y = 0.0f;

    // C: bias broadcast into all accumulator slots
    const float bv = bias[f];
    v8f c;
    #pragma unroll
    for (int i = 0; i < 8; ++i) c[i] = bv;

    // D = A x B + C  (EXEC is all 1s here: whole wave executes unconditionally)
    v8f d = __builtin_amdgcn_wmma_f32_16x16x4_f32(
        /*neg_a=*/false, a, /*neg_b=*/false, bmat,
        /*c_mod=*/(short)0, c, /*reuse_a=*/false, /*reuse_b=*/false);

    // D layout: lanes 0-15 -> N=lane, M=r; lanes 16-31 -> N=lane-16, M=8+r
    if (valid) {
        const unsigned base = (unsigned)f * (NROWS * PNUM);
        const int m0 = lo ? 0 : 8;
        #pragma unroll
        for (int r = 0; r < 8; ++r) {
            float v = d[r];
            v = v > 0.0f ? v : 0.0f;                    // ReLU
            const unsigned idx = base + (unsigned)(nb + m0 + r) * PNUM + pcol;
            __builtin_nontemporal_store(v, out + idx);  // write-once 210MB stream
        }
    }
}

// ---------------------------------------------------------------------------
extern "C" void kernel_launch(void* const* d_in, const int* in_sizes, int n_in,
                              void* d_out, int out_size, void* d_ws, size_t ws_size,
                              hipStream_t stream) {
    const float* x  = (const float*)d_in[0];   // (8,64,64,64) f32
    const float* W  = (const float*)d_in[1];   // (64,5,5) f32 -> (64,25) flat
    const float* b  = (const float*)d_in[2];   // (64,) f32
    float*       o  = (float*)d_out;           // (1,64,32768,25) f32
    float*       s  = (float*)d_ws;            // 32768 floats scratch (128 KB)

    // Kernel 1: 32768 waves, 8 waves per 256-thread block -> 4096 blocks
    row_sum_kernel<<<NROWS / 8, 256, 0, stream>>>(x, s);

    // Kernel 2: 64 f * 2048 n-tiles * 2 p-tiles = 262144 waves -> 32768 blocks
    outer_wmma_kernel<<<(FNUM * 2048 * 2) / 8, 256, 0, stream>>>(s, W, b, o);
}